// SimpleGraphNet_78881369358664
// MI455X (gfx1250) — compile-verified
//
#include <hip/hip_runtime.h>

#define N_NODES  100000
#define N_EDGES  3200000
#define N_GRAPHS 2048
#define HDIM     64
#define N_LAYERS 3

typedef __attribute__((ext_vector_type(2))) float v2f;
typedef __attribute__((ext_vector_type(8))) float v8f;

// ---------------- utility fills ----------------
__global__ void fill_f32(float* __restrict__ p, float v, int n) {
  int i = blockIdx.x * blockDim.x + threadIdx.x;
  if (i < n) p[i] = v;
}

// ---------------- degree / norm ----------------
__global__ void deg_kernel(const int* __restrict__ col, float* __restrict__ deg, int n_edges) {
  int e = blockIdx.x * blockDim.x + threadIdx.x;
  if (e < n_edges) atomicAdd(&deg[col[e]], 1.0f);
}

__global__ void rsqrt_kernel(float* __restrict__ deg, int n) {
  int i = blockIdx.x * blockDim.x + threadIdx.x;
  if (i < n) deg[i] = rsqrtf(deg[i]);   // deg >= 1 always (self-loop)
}

// ---------------- embedding gather ----------------
__global__ void embed_kernel(const int* __restrict__ x, const float* __restrict__ emb,
                             float* __restrict__ h, int n_elems) {
  int i = blockIdx.x * blockDim.x + threadIdx.x;
  if (i >= n_elems) return;
  int node = i >> 6;
  int j = i & 63;
  h[i] = emb[x[node] * HDIM + j];
}

// ---------------- WMMA f32 GEMM: t[16n x 64] = h[16n x 64] @ W[64 x 64] ----------------
// One block = 128 threads = 4 waves; block handles a 16-node tile; wave w owns
// output columns [16w, 16w+16). K accumulated in steps of 4 via V_WMMA_F32_16X16X4_F32.
__global__ __launch_bounds__(128) void gemm_wmma(const float* __restrict__ h,
                                                 const float* __restrict__ W,
                                                 float* __restrict__ t) {
  __shared__ float sW[HDIM * HDIM];   // 16 KB
  __shared__ float sH[16 * HDIM];     // 4 KB
  const int tid  = threadIdx.x;
  const int tile = blockIdx.x;

  for (int i = tid; i < HDIM * HDIM; i += 128) sW[i] = W[i];
  const float* hbase = h + (size_t)tile * 16 * HDIM;
  for (int i = tid; i < 16 * HDIM; i += 128) sH[i] = hbase[i];
  __syncthreads();

  const int wave = tid >> 5;
  const int lane = tid & 31;
  const int half = lane >> 4;     // 0: lanes 0-15, 1: lanes 16-31
  const int l16  = lane & 15;
  const int colb = wave * 16;

  v8f acc = {};
#pragma unroll
  for (int k = 0; k < HDIM; k += 4) {
    const int ka = k + 2 * half;
    v2f a, b;
    // A 16x4 layout: lane<16 -> row=l16, K = k..k+1 ; lane>=16 -> row=l16, K = k+2..k+3
    a.x = sH[l16 * HDIM + ka];
    a.y = sH[l16 * HDIM + ka + 1];
    // B 4x16 layout: rows striped across lanes; rows k,k+1 in lanes 0-15, rows k+2,k+3 in lanes 16-31
    b.x = sW[ka * HDIM + colb + l16];
    b.y = sW[(ka + 1) * HDIM + colb + l16];
    acc = __builtin_amdgcn_wmma_f32_16x16x4_f32(false, a, false, b, (short)0, acc, false, false);
  }

  // D 16x16: VGPR i holds row (half*8 + i), col = l16
  float* tbase = t + ((size_t)tile * 16 + half * 8) * HDIM + colb + l16;
#pragma unroll
  for (int i = 0; i < 8; i++) tbase[i * HDIM] = acc[i];
}

// ---------------- edge scatter: agg[col] += t[row] * (dinv[row]*dinv[col]) ----------------
// One wave per edge; lane handles features {lane, lane+32}.
__global__ void scatter_kernel(const int* __restrict__ row, const int* __restrict__ col,
                               const float* __restrict__ dinv, const float* __restrict__ t,
                               float* __restrict__ agg, int n_edges) {
  unsigned long long gid = (unsigned long long)blockIdx.x * blockDim.x + threadIdx.x;
  int e = (int)(gid >> 5);
  int lane = (int)(gid & 31);
  if (e >= n_edges) return;
  int r = row[e];
  int c = col[e];
  float f = dinv[r] * dinv[c];
  const float* ts = t + (size_t)r * HDIM;
  float* as = agg + (size_t)c * HDIM;
  atomicAdd(&as[lane],      ts[lane]      * f);
  atomicAdd(&as[lane + 32], ts[lane + 32] * f);
}

// ---------------- self-loop + bias + relu + residual ----------------
__global__ void relu_update(const float* __restrict__ t, const float* __restrict__ agg,
                            const float* __restrict__ dinv, const float* __restrict__ bias,
                            float* __restrict__ h, int layer, int n_elems) {
  int i = blockIdx.x * blockDim.x + threadIdx.x;
  if (i >= n_elems) return;
  int node = i >> 6;
  int j = i & 63;
  float d = dinv[node];
  float v = agg[i] + t[i] * d * d + bias[j];
  v = fmaxf(v, 0.0f);
  h[i] = (layer == 0) ? v : (h[i] + v);
}

// ---------------- pooling ----------------
__global__ void pool_sum(const float* __restrict__ h, const int* __restrict__ batch,
                         float* __restrict__ gsum, int n_elems) {
  int i = blockIdx.x * blockDim.x + threadIdx.x;
  if (i >= n_elems) return;
  int node = i >> 6;
  int j = i & 63;
  atomicAdd(&gsum[batch[node] * HDIM + j], h[i]);
}

__global__ void pool_cnt(const int* __restrict__ batch, float* __restrict__ cnt, int n) {
  int i = blockIdx.x * blockDim.x + threadIdx.x;
  if (i < n) atomicAdd(&cnt[batch[i]], 1.0f);
}

// ---------------- predictor MLP: 64 -> 32 -> 16 -> 1, one block of 64 threads per graph ----
__global__ __launch_bounds__(64) void mlp_kernel(const float* __restrict__ gsum,
                                                 const float* __restrict__ cnt,
                                                 const float* __restrict__ w1, const float* __restrict__ b1,
                                                 const float* __restrict__ w2, const float* __restrict__ b2,
                                                 const float* __restrict__ w3, const float* __restrict__ b3,
                                                 float* __restrict__ out) {
  __shared__ float sg[64];
  __shared__ float s1[32];
  __shared__ float s2[16];
  int g = blockIdx.x;
  int tIdx = threadIdx.x;
  float inv = 1.0f / fmaxf(cnt[g], 1.0f);
  sg[tIdx] = gsum[g * HDIM + tIdx] * inv;
  __syncthreads();
  if (tIdx < 32) {
    float acc = b1[tIdx];
#pragma unroll
    for (int k = 0; k < 64; k++) acc += sg[k] * w1[k * 32 + tIdx];
    s1[tIdx] = fmaxf(acc, 0.0f);
  }
  __syncthreads();
  if (tIdx < 16) {
    float acc = b2[tIdx];
#pragma unroll
    for (int k = 0; k < 32; k++) acc += s1[k] * w2[k * 16 + tIdx];
    s2[tIdx] = fmaxf(acc, 0.0f);
  }
  __syncthreads();
  if (tIdx == 0) {
    float acc = b3[0];
#pragma unroll
    for (int k = 0; k < 16; k++) acc += s2[k] * w3[k];
    out[g] = acc;
  }
}

// ---------------- host-side orchestration ----------------
static inline size_t align256(size_t v) { return (v + 255) & ~(size_t)255; }

extern "C" void kernel_launch(void* const* d_in, const int* in_sizes, int n_in,
                              void* d_out, int out_size, void* d_ws, size_t ws_size,
                              hipStream_t stream) {
  (void)in_sizes; (void)n_in; (void)out_size; (void)ws_size;

  const int*   x      = (const int*)d_in[0];
  const int*   edge   = (const int*)d_in[1];     // [2, E] flat
  const int*   batch  = (const int*)d_in[2];
  const float* emb    = (const float*)d_in[3];   // [92, 64]
  const float* conv_w = (const float*)d_in[4];   // [3, 64, 64]
  const float* conv_b = (const float*)d_in[5];   // [3, 64]
  const float* w1     = (const float*)d_in[6];   // [64, 32]
  const float* b1     = (const float*)d_in[7];
  const float* w2     = (const float*)d_in[8];   // [32, 16]
  const float* b2     = (const float*)d_in[9];
  const float* w3     = (const float*)d_in[10];  // [16, 1]
  const float* b3     = (const float*)d_in[11];
  float* out = (float*)d_out;

  const int* erow = edge;            // edge_index[0] = source (gather)
  const int* ecol = edge + N_EDGES;  // edge_index[1] = target (scatter)

  // workspace carve-out
  char* ws = (char*)d_ws;
  float* dinv = (float*)ws;  ws += align256((size_t)N_NODES * 4);
  float* h    = (float*)ws;  ws += align256((size_t)N_NODES * HDIM * 4);
  float* t    = (float*)ws;  ws += align256((size_t)N_NODES * HDIM * 4);
  float* agg  = (float*)ws;  ws += align256((size_t)N_NODES * HDIM * 4);
  float* gsum = (float*)ws;  ws += align256((size_t)N_GRAPHS * HDIM * 4);
  float* cnt  = (float*)ws;  ws += align256((size_t)N_GRAPHS * 4);

  const int NE = N_NODES * HDIM;     // 6,400,000

  // 1) symmetric degree normalization (in-degree + self-loop)
  fill_f32<<<(N_NODES + 255) / 256, 256, 0, stream>>>(dinv, 1.0f, N_NODES);
  deg_kernel<<<(N_EDGES + 255) / 256, 256, 0, stream>>>(ecol, dinv, N_EDGES);
  rsqrt_kernel<<<(N_NODES + 255) / 256, 256, 0, stream>>>(dinv, N_NODES);

  // 2) embedding
  embed_kernel<<<(NE + 255) / 256, 256, 0, stream>>>(x, emb, h, NE);

  // 3) GCN layers
  for (int l = 0; l < N_LAYERS; l++) {
    gemm_wmma<<<N_NODES / 16, 128, 0, stream>>>(h, conv_w + (size_t)l * HDIM * HDIM, t);
    fill_f32<<<(NE + 255) / 256, 256, 0, stream>>>(agg, 0.0f, NE);
    scatter_kernel<<<((size_t)N_EDGES * 32 + 255) / 256, 256, 0, stream>>>(erow, ecol, dinv, t, agg, N_EDGES);
    relu_update<<<(NE + 255) / 256, 256, 0, stream>>>(t, agg, dinv, conv_b + (size_t)l * HDIM, h, l, NE);
  }

  // 4) global mean pool
  fill_f32<<<(N_GRAPHS * HDIM + 255) / 256, 256, 0, stream>>>(gsum, 0.0f, N_GRAPHS * HDIM);
  fill_f32<<<(N_GRAPHS + 255) / 256, 256, 0, stream>>>(cnt, 0.0f, N_GRAPHS);
  pool_sum<<<(NE + 255) / 256, 256, 0, stream>>>(h, batch, gsum, NE);
  pool_cnt<<<(N_NODES + 255) / 256, 256, 0, stream>>>(batch, cnt, N_NODES);

  // 5) predictor MLP
  mlp_kernel<<<N_GRAPHS, 64, 0, stream>>>(gsum, cnt, w1, b1, w2, b2, w3, b3, out);
}